// Unitary_4217657884878
// MI455X (gfx1250) — compile-verified
//
#include <hip/hip_runtime.h>

#define NM 512   // N_MODES; K = 511 thetas

typedef float v2f __attribute__((ext_vector_type(2)));
typedef float v8f __attribute__((ext_vector_type(8)));

// Kernel 1: cs[k] = cos(theta[k]), sn[k] = sin(theta[k]) for k < K;
// pad cs[K..511] = 1, sn[K..511] = 0 so that cos'_{511} == 1 falls out naturally.
__global__ __launch_bounds__(NM) void unitary_trig_kernel(
    const float* __restrict__ th, float* __restrict__ cs,
    float* __restrict__ sn, int K) {
  int k = threadIdx.x;
  if (k >= NM) return;
  if (k < K) {
    float t = th[k];
    cs[k] = cosf(t);
    sn[k] = sinf(t);
  } else {
    cs[k] = 1.0f;
    sn[k] = 0.0f;
  }
}

// Kernel 2: one wave (32 lanes) per 16-column stripe. Closed-form fill:
//   U[m,j] = 0                         (m < j-1)
//   U[j-1,j] = sn[j-1]
//   U[m,j] = b_j * cs'[m] * t_{j,m}    (m >= j),  t_{j,j}=1, t_{j,m}=-sn[m-1]*t_{j,m-1}
// Tiles strictly below the diagonal band are rank-1 and computed with
// V_WMMA_F32_16X16X4_F32 (outer product via K=0 slice), full fp32 precision.
__global__ __launch_bounds__(32) void unitary_fill_kernel(
    const float* __restrict__ g_cs, const float* __restrict__ g_sn,
    float* __restrict__ out) {
  __shared__ float CS[NM];
  __shared__ float SN[NM];

  const int lane = threadIdx.x;
  for (int i = lane; i < NM; i += 32) {
    CS[i] = g_cs[i];
    SN[i] = g_sn[i];
  }
  __syncthreads();

  const int Jblk = blockIdx.x;      // 0..31
  const int J0   = Jblk << 4;       // first column of this stripe
  const int l15  = lane & 15;
  const int j    = J0 + l15;        // column owned by lanes 0..15
  const bool lo  = (lane < 16);
  const float b_j = lo ? ((j == 0) ? 1.0f : CS[j - 1]) : 0.0f;

  float tcol = 0.0f;  // t_{j, last-processed-row}; initialized in the band chunk

  for (int ci = 0; ci < 32; ++ci) {
    const int M = ci << 4;          // first row of this 16-row chunk (uniform)

    if (ci > Jblk) {
      // -------- strictly-below tile: rank-1 outer product via WMMA --------
      // advance t to row M (last processed row was M-1)
      const float tM = -SN[M - 1] * tcol;
      float u = 0.0f, v = 0.0f;
      if (lo) {
        // q_r = (-1)^r * prod_{l=M}^{M+r-1} sn[l]   (r = l15)
        float q = 1.0f;
        for (int i = 0; i < l15; ++i) q = -q * SN[M + i];
        u = CS[M + l15] * q;   // row factor  a_m * q_m   (CS[511]==1 pads last row)
        v = b_j * tM;          // col factor  b_j * t_{j,M}
      }
      // A: 16x4 fp32, only K=0 column non-zero (VGPR0, lanes 0-15).
      // B: 4x16 fp32, only K=0 row non-zero (VGPR0, lanes 0-15).
      v2f A; A.x = u; A.y = 0.0f;
      v2f B; B.x = v; B.y = 0.0f;
      v8f C = {};
      v8f D = __builtin_amdgcn_wmma_f32_16x16x4_f32(
          false, A, false, B, (short)0, C, false, false);
      // D layout: lane%16 = N, VGPR r = row r (lanes 0-15) / row 8+r (lanes 16-31)
      const int rowbase = M + ((lane >> 4) << 3);
      #pragma unroll
      for (int r = 0; r < 8; ++r)
        out[(rowbase + r) * NM + J0 + l15] = D[r];
      // advance t to row M+15:  t_{M+15} = t_M * (-1)^15 * prod_{l=M}^{M+14} sn[l]
      float p = 1.0f;
      #pragma unroll
      for (int i = 0; i < 15; ++i) p *= SN[M + i];
      tcol = -tM * p;
    } else if (ci >= Jblk - 1) {
      // -------- diagonal-band chunks (ci == Jblk-1, Jblk): scalar recurrence ----
      for (int r = 0; r < 16; ++r) {
        const int m = M + r;
        if (lo) {
          float val = 0.0f;
          if (m + 1 == j) {
            val = SN[m];                        // superdiagonal
          } else if (m == j) {
            tcol = 1.0f;                        // recurrence start
            val = b_j * CS[m];                  // diagonal
          } else if (m > j) {
            tcol = -SN[m - 1] * tcol;
            val = b_j * CS[m] * tcol;
          }
          out[m * NM + j] = val;                // zeros above band written too
        }
      }
    } else {
      // -------- strictly-above tile: zeros --------
      const int rowbase = M + ((lane >> 4) << 3);
      #pragma unroll
      for (int r = 0; r < 8; ++r)
        out[(rowbase + r) * NM + J0 + l15] = 0.0f;
    }
  }
}

extern "C" void kernel_launch(void* const* d_in, const int* in_sizes, int n_in,
                              void* d_out, int out_size, void* d_ws, size_t ws_size,
                              hipStream_t stream) {
  (void)n_in; (void)out_size; (void)ws_size;
  const float* thetas = (const float*)d_in[0];
  const int K = in_sizes[0];                 // 511
  float* cs = (float*)d_ws;                  // 512 floats
  float* sn = cs + NM;                       // 512 floats  (4 KB total scratch)

  unitary_trig_kernel<<<1, NM, 0, stream>>>(thetas, cs, sn, K);
  unitary_fill_kernel<<<NM / 16, 32, 0, stream>>>(cs, sn, (float*)d_out);
}